// GINEBlock_45397804319446
// MI455X (gfx1250) — compile-verified
//
#include <hip/hip_runtime.h>

typedef __attribute__((ext_vector_type(2))) float v2f;
typedef __attribute__((ext_vector_type(8))) float v8f;

#define HIDDEN 128
#define NTYPES 16
#define EMBD 8
#define MLP_BLOCKS 512

// ---------------------------------------------------------------------------
// Kernel 1: proj[t][f] = sum_k emb[t][k] * We[k][f] + be[f]   (16 x 128 table)
// ---------------------------------------------------------------------------
__global__ void gine_proj_kernel(const float* __restrict__ emb,
                                 const float* __restrict__ We,
                                 const float* __restrict__ be,
                                 float* __restrict__ proj) {
    int idx = blockIdx.x * blockDim.x + threadIdx.x;   // 0..2047
    int t = idx >> 7;
    int f = idx & (HIDDEN - 1);
    float s = be[f];
#pragma unroll
    for (int k = 0; k < EMBD; ++k)
        s += emb[t * EMBD + k] * We[k * HIDDEN + f];
    proj[idx] = s;
}

// ---------------------------------------------------------------------------
// Kernel 2: z = (1 + eps) * x   (accumulator init, written into d_out)
// ---------------------------------------------------------------------------
__global__ void gine_init_kernel(const float* __restrict__ x,
                                 const float* __restrict__ eps,
                                 float* __restrict__ z) {
    int idx = blockIdx.x * blockDim.x + threadIdx.x;   // float4 index
    float s = 1.0f + eps[0];
    float4 v = ((const float4*)x)[idx];
    v.x *= s; v.y *= s; v.z *= s; v.w *= s;
    ((float4*)z)[idx] = v;
}

// ---------------------------------------------------------------------------
// Kernel 3: scatter-sum of relu(x[src] + proj[h]) into z[dst] via f32 atomics
// One wave (32 lanes x float4 = 128 floats) per edge; 8 edges per block.
// Accumulator (25.6 MB) is L2-resident (192 MB L2) so atomics RMW in L2.
// ---------------------------------------------------------------------------
__global__ void gine_scatter_kernel(const float* __restrict__ x,
                                    const int* __restrict__ edge_index,
                                    const int* __restrict__ edge_attr,
                                    const float* __restrict__ proj,
                                    float* __restrict__ z, int E) {
    __shared__ float sproj[NTYPES * HIDDEN];
    int tid = threadIdx.x;
    for (int i = tid; i < NTYPES * HIDDEN; i += 256) sproj[i] = proj[i];
    __syncthreads();

    int wave = tid >> 5;
    int lane = tid & 31;
    int e = blockIdx.x * 8 + wave;
    if (e >= E) return;

    int src = edge_index[e];
    int dst = edge_index[E + e];
    int a0 = edge_attr[3 * e + 0];
    int a1 = edge_attr[3 * e + 1];
    int a2 = edge_attr[3 * e + 2];
    int h = (a0 + 3 * a1 + 7 * a2) & (NTYPES - 1);

    int f = lane * 4;
    const float4 xv = *(const float4*)(x + (size_t)src * HIDDEN + f);
    const float4 pv = *(const float4*)(sproj + h * HIDDEN + f);
    float4 m;
    m.x = fmaxf(xv.x + pv.x, 0.0f);
    m.y = fmaxf(xv.y + pv.y, 0.0f);
    m.z = fmaxf(xv.z + pv.z, 0.0f);
    m.w = fmaxf(xv.w + pv.w, 0.0f);

    float* zp = z + (size_t)dst * HIDDEN + f;
    atomicAdd(zp + 0, m.x);
    atomicAdd(zp + 1, m.y);
    atomicAdd(zp + 2, m.z);
    atomicAdd(zp + 3, m.w);
}

// ---------------------------------------------------------------------------
// Kernel 4: out = relu(z @ W1 + b1) @ W2 + b2, fp32 WMMA 16x16x4.
// Persistent blocks: stage W1+W2 in LDS once (coalesced b128), then
// grid-stride over 16-row M-tiles. 8 waves = 8 N-tiles of 16. In-place on
// d_out: each block reads its tile into LDS before writing those rows back.
// ---------------------------------------------------------------------------
__global__ void gine_mlp_kernel(float* __restrict__ zout,
                                const float* __restrict__ W1,
                                const float* __restrict__ b1,
                                const float* __restrict__ W2,
                                const float* __restrict__ b2,
                                int ntiles) {
    __shared__ float ldsW1[HIDDEN * HIDDEN];   // 64 KB
    __shared__ float ldsW2[HIDDEN * HIDDEN];   // 64 KB
    __shared__ float lds_z[16 * HIDDEN];       // 8 KB
    __shared__ float lds_h[16 * HIDDEN];       // 8 KB

    int tid = threadIdx.x;

    // Stage both weight matrices once: 4096 float4 each, 16 per thread.
#pragma unroll
    for (int i = 0; i < 16; ++i) {
        int idx4 = tid + i * 256;
        ((float4*)ldsW1)[idx4] = ((const float4*)W1)[idx4];
        ((float4*)ldsW2)[idx4] = ((const float4*)W2)[idx4];
    }

    int wave = tid >> 5;
    int lane = tid & 31;
    int n0 = wave * 16;
    int half = (lane >= 16) ? 1 : 0;  // lane half selects K pair (A/B frags)
    int m  = lane & 15;               // local M for A fragment
    int nl = lane & 15;               // local N for B fragment
    int col = n0 + nl;
    float bias1 = b1[col];            // per-lane constants across all tiles
    float bias2 = b2[col];

    __syncthreads();

    for (int tile = blockIdx.x; tile < ntiles; tile += gridDim.x) {
        int rowBase = tile * 16;
        const float* ztile = zout + (size_t)rowBase * HIDDEN;

        // Prefetch this block's next tile while computing the current one.
        int ntile = tile + gridDim.x;
        if (ntile < ntiles) {
            const char* np = (const char*)(zout + (size_t)ntile * 16 * HIDDEN);
            __builtin_prefetch(np + tid * 32, 0, 0);   // 8 KB, 32 B per thread
        }

        // Cooperative load of the 16x128 activation tile (512 float4).
#pragma unroll
        for (int i = 0; i < 2; ++i) {
            int idx4 = tid + i * 256;
            ((float4*)lds_z)[idx4] = ((const float4*)ztile)[idx4];
        }
        __syncthreads();

        // ---- GEMM1: h1 = z @ W1 ----
        v8f acc = {};
        for (int k0 = 0; k0 < HIDDEN; k0 += 4) {
            int ka = k0 + half * 2;
            v2f a; a.x = lds_z[m * HIDDEN + ka];
                   a.y = lds_z[m * HIDDEN + ka + 1];
            v2f b; b.x = ldsW1[ka * HIDDEN + col];
                   b.y = ldsW1[(ka + 1) * HIDDEN + col];
            acc = __builtin_amdgcn_wmma_f32_16x16x4_f32(
                false, a, false, b, (short)0, acc, false, false);
        }

        // bias + relu, share through LDS (C/D layout: VGPR r -> M = r+half*8)
#pragma unroll
        for (int r = 0; r < 8; ++r) {
            int M = half * 8 + r;
            lds_h[M * HIDDEN + col] = fmaxf(acc[r] + bias1, 0.0f);
        }
        __syncthreads();

        // ---- GEMM2: out = h1 @ W2 ----
        v8f acc2 = {};
        for (int k0 = 0; k0 < HIDDEN; k0 += 4) {
            int ka = k0 + half * 2;
            v2f a; a.x = lds_h[m * HIDDEN + ka];
                   a.y = lds_h[m * HIDDEN + ka + 1];
            v2f b; b.x = ldsW2[ka * HIDDEN + col];
                   b.y = ldsW2[(ka + 1) * HIDDEN + col];
            acc2 = __builtin_amdgcn_wmma_f32_16x16x4_f32(
                false, a, false, b, (short)0, acc2, false, false);
        }

#pragma unroll
        for (int r = 0; r < 8; ++r) {
            int M = half * 8 + r;
            zout[(size_t)(rowBase + M) * HIDDEN + col] = acc2[r] + bias2;
        }
        __syncthreads();   // protect lds_z/lds_h before next iteration
    }
}

// ---------------------------------------------------------------------------
extern "C" void kernel_launch(void* const* d_in, const int* in_sizes, int n_in,
                              void* d_out, int out_size, void* d_ws, size_t ws_size,
                              hipStream_t stream) {
    const float* x    = (const float*)d_in[0];
    const int*   eidx = (const int*)d_in[1];
    const int*   eatt = (const int*)d_in[2];
    const float* emb  = (const float*)d_in[3];
    const float* We   = (const float*)d_in[4];
    const float* be   = (const float*)d_in[5];
    const float* W1   = (const float*)d_in[6];
    const float* b1   = (const float*)d_in[7];
    const float* W2   = (const float*)d_in[8];
    const float* b2   = (const float*)d_in[9];
    const float* eps  = (const float*)d_in[10];
    float* out  = (float*)d_out;
    float* proj = (float*)d_ws;          // 16*128 floats = 8 KB scratch

    const int N = in_sizes[0] / HIDDEN;  // 50000
    const int E = in_sizes[1] / 2;       // 640000
    const int ntiles = (N + 15) / 16;    // 3125

    // 1) 16x128 edge-type projection table
    gine_proj_kernel<<<(NTYPES * HIDDEN) / 256, 256, 0, stream>>>(emb, We, be, proj);
    // 2) accumulator init: out = (1+eps)*x   (float4 elements)
    gine_init_kernel<<<(N * HIDDEN / 4 + 255) / 256, 256, 0, stream>>>(x, eps, out);
    // 3) edge scatter with f32 atomics (8 edges / 256-thread block)
    gine_scatter_kernel<<<(E + 7) / 8, 256, 0, stream>>>(x, eidx, eatt, proj, out, E);
    // 4) persistent-block fused 2-layer MLP with fp32 WMMA, in place on d_out
    int mlpBlocks = (ntiles < MLP_BLOCKS) ? ntiles : MLP_BLOCKS;
    gine_mlp_kernel<<<mlpBlocks, 256, 0, stream>>>(out, W1, b1, W2, b2, ntiles);
}